// QueryAndGroup_87686052315598
// MI455X (gfx1250) — compile-verified
//
#include <hip/hip_runtime.h>
#include <stdint.h>

// Problem constants (from reference setup_inputs)
#define B_  4
#define N_  16384
#define S_  2048
#define C_  64
#define K_  64          // nsample
#define R2_ 0.01f       // radius^2

typedef __attribute__((ext_vector_type(2))) float v2f;
typedef __attribute__((ext_vector_type(4))) float v4f;
typedef __attribute__((ext_vector_type(8))) float v8f;

// ---------------------------------------------------------------------------
// Kernel 1: ball query.
//   grid  = B_ * (S_/128) = 64 blocks, 256 threads (8 waves).
//   Each block: async-stage xyz[b] (128 KB) into LDS, then each wave handles
//   16 queries using V_WMMA_F32_16X16X4_F32 to produce 16x16 distance tiles,
//   ballots -> per-query in-order index extraction with early exit at 64.
// ---------------------------------------------------------------------------
__global__ __launch_bounds__(256) void bq_kernel(const float* __restrict__ xyz,
                                                 const float* __restrict__ new_xyz,
                                                 const int*  __restrict__ pointsnum,
                                                 int*        __restrict__ idx_out)
{
    __shared__ float2 pts[N_];   // 128 KB < 320 KB WGP LDS

    const int tid  = threadIdx.x;
    const int lane = tid & 31;
    const int wave = tid >> 5;
    const int blocks_per_batch = S_ / 128;
    const int b     = blockIdx.x / blocks_per_batch;
    const int qbase = (blockIdx.x % blocks_per_batch) * 128;

    // ---- async stage xyz[b] (N_ * float2 = 128 KB) into LDS ----
    // per wave: 16 KB via 32 x b128 ops (16 B per lane per op)
    {
        const char*   gsrc     = (const char*)(xyz + (size_t)b * N_ * 2);
        const uint32_t lds_base = (uint32_t)(uintptr_t)(&pts[0]);
        const int base = wave * 16384 + lane * 16;
        for (int t = 0; t < 32; ++t) {
            uint32_t    loff = lds_base + (uint32_t)(base + t * 512);
            const char* g    = gsrc + base + t * 512;
            asm volatile("global_load_async_to_lds_b128 %0, %1, off"
                         :: "v"(loff), "v"(g) : "memory");
        }
        asm volatile("s_wait_asynccnt 0" ::: "memory");
    }
    __syncthreads();

    const int pnum = min(pointsnum[b], N_);

    // ---- per-wave: 16 queries. Build A (16x4): row m = (qx,qy,0,0) ----
    const int q = qbase + wave * 16 + (lane & 15);
    v2f   A = {0.f, 0.f};
    float qn_own = 0.f;
    if (lane < 16) {
        float2 c = ((const float2*)new_xyz)[(size_t)b * S_ + q];
        A.x = c.x; A.y = c.y;
        qn_own = c.x * c.x + c.y * c.y;
    }
    // qn8[r] = |q|^2 of query m = r + 8*(lane>=16)  (matches D lane layout)
    const int hi = (lane >= 16) ? 8 : 0;
    float qn8[8];
#pragma unroll
    for (int r = 0; r < 8; ++r) qn8[r] = __shfl(qn_own, r + hi, 32);

    int  cnt    = (lane < 16) ? 0 : 64;   // lanes 16..31: permanently "done"
    int  firsti = 0;
    bool any    = false;
    int* myout  = idx_out + ((size_t)b * S_ + q) * K_;

    const int nchunks = (pnum + 15) >> 4;
    for (int ch = 0; ch < nchunks; ++ch) {
        // wave-uniform early exit -> EXEC all-ones at the WMMA
        if (__builtin_amdgcn_ballot_w32(cnt >= 64) == 0xFFFFFFFFu) break;

        const int p0 = ch << 4;
        float2 p  = pts[p0 + (lane & 15)];
        float  pn = p.x * p.x + p.y * p.y;
        v2f Bv = {0.f, 0.f};
        if (lane < 16) { Bv.x = p.x; Bv.y = p.y; }  // cols K=0,1; K=2,3 zero

        v8f cacc = {};
        v8f d = __builtin_amdgcn_wmma_f32_16x16x4_f32(
                    false, A, false, Bv, (short)0, cacc, false, false);

        const bool validn = (p0 + (lane & 15)) < pnum;
        uint32_t bal[8];
#pragma unroll
        for (int r = 0; r < 8; ++r) {
            float d2 = qn8[r] + pn - 2.0f * d[r];     // |q-p|^2
            bal[r] = __builtin_amdgcn_ballot_w32(validn && (d2 < R2_));
        }
        // lane q (<16) owns query q: pick ballot r=q&7, half = (q<8) ? lo : hi
        const int l3 = lane & 7;
        uint32_t t0 = (l3 & 1) ? bal[1] : bal[0];
        uint32_t t1 = (l3 & 1) ? bal[3] : bal[2];
        uint32_t t2 = (l3 & 1) ? bal[5] : bal[4];
        uint32_t t3 = (l3 & 1) ? bal[7] : bal[6];
        uint32_t u0 = (l3 & 2) ? t1 : t0;
        uint32_t u1 = (l3 & 2) ? t3 : t2;
        uint32_t bq = (l3 & 4) ? u1 : u0;
        if (lane < 16 && cnt < 64) {
            uint32_t m16 = (lane < 8) ? (bq & 0xFFFFu) : (bq >> 16);
            while (m16 && cnt < 64) {
                int n = __builtin_ctz(m16);
                m16 &= m16 - 1;
                int pi = p0 + n;
                if (!any) { firsti = pi; any = true; }
                myout[cnt++] = pi;
            }
        }
    }

    // pad remaining slots with first found index (0 if none)
    if (lane < 16) {
        int pad = any ? firsti : 0;
        for (int k = cnt; k < K_; ++k) myout[k] = pad;
    }
}

// ---------------------------------------------------------------------------
// Kernel 2: gather + concat.  out[b][c][s][k], c in [0,66):
//   c<2 : xyz[b][idx][c] - new_xyz[b][s][c]
//   c>=2: features[b][c-2][idx]
// grid = B_*S_ blocks, 256 threads. Each thread owns 4 consecutive k and one
// channel group: per channel it does 4 gathered loads (L2-resident sources)
// and ONE aligned float4 NON-TEMPORAL store (write-once 138 MB stream must
// not evict the 16 MB gather working set from L2).
// ---------------------------------------------------------------------------
__global__ __launch_bounds__(256) void group_kernel(const float* __restrict__ xyz,
                                                    const float* __restrict__ new_xyz,
                                                    const float* __restrict__ feat,
                                                    const int*  __restrict__ idx,
                                                    float*      __restrict__ out)
{
    __shared__ int   sidx[K_];
    __shared__ float ctr[2];
    const int b   = blockIdx.x / S_;
    const int s   = blockIdx.x % S_;
    const int tid = threadIdx.x;
    if (tid < K_) sidx[tid] = idx[((size_t)b * S_ + s) * K_ + tid];
    if (tid < 2)  ctr[tid]  = new_xyz[((size_t)b * S_ + s) * 2 + tid];
    __syncthreads();

    const int kq = (tid & 15) * 4;    // 4 consecutive k per thread
    const int cg = tid >> 4;          // 0..15 channel group
    int id0 = sidx[kq + 0];
    int id1 = sidx[kq + 1];
    int id2 = sidx[kq + 2];
    int id3 = sidx[kq + 3];

    const float* xb = xyz  + (size_t)b * N_ * 2;
    const float* fb = feat + (size_t)b * C_ * N_;
    const size_t outbase = (((size_t)b * 66) * S_ + s) * K_ + kq;

    for (int c = cg; c < 66; c += 16) {
        v4f v;
        if (c < 2) {
            float ctr_c = ctr[c];
            v.x = xb[id0 * 2 + c] - ctr_c;
            v.y = xb[id1 * 2 + c] - ctr_c;
            v.z = xb[id2 * 2 + c] - ctr_c;
            v.w = xb[id3 * 2 + c] - ctr_c;
        } else {
            const float* row = fb + (size_t)(c - 2) * N_;
            v.x = row[id0];
            v.y = row[id1];
            v.z = row[id2];
            v.w = row[id3];
        }
        __builtin_nontemporal_store(v, (v4f*)(out + outbase + (size_t)c * (S_ * K_)));
    }
}

// ---------------------------------------------------------------------------
extern "C" void kernel_launch(void* const* d_in, const int* in_sizes, int n_in,
                              void* d_out, int out_size, void* d_ws, size_t ws_size,
                              hipStream_t stream) {
    (void)in_sizes; (void)n_in; (void)out_size; (void)ws_size;
    const float* xyz       = (const float*)d_in[0];
    const float* new_xyz   = (const float*)d_in[1];
    const float* feat      = (const float*)d_in[2];
    const int*   pointsnum = (const int*)d_in[3];
    float* out    = (float*)d_out;
    int*   idx_ws = (int*)d_ws;   // needs B_*S_*K_*4 = 2 MB of scratch

    bq_kernel<<<B_ * (S_ / 128), 256, 0, stream>>>(xyz, new_xyz, pointsnum, idx_ws);
    group_kernel<<<B_ * S_, 256, 0, stream>>>(xyz, new_xyz, feat, idx_ws, out);
}